// DepthwiseCorrelation_21122649162559
// MI455X (gfx1250) — compile-verified
//
#include <hip/hip_runtime.h>
#include <hip/hip_bf16.h>

// ---------------------------------------------------------------------------
// MI455X (gfx1250) implementation of DepthwiseCorrelation.
// GEMM-shaped work (both 1x1 convs-as-GEMM and the 3x3 conv as implicit GEMM)
// runs on V_WMMA_F32_16X16X4_F32 (exact fp32 matrix path, wave32 tiles).
// Depthwise 7x7 correlation runs on VALU FMA out of an LDS halo tile.
// All boundary handling uses clamped addresses + v_cndmask selects so every
// global load in the WMMA loops is unconditional (no EXEC-mask branching).
// Intermediates live in d_ws (2 x 32 MB) and stay resident in the 192 MB L2,
// so im2col re-reads for the 3x3 conv are L2 hits.
// ---------------------------------------------------------------------------

typedef float v2f __attribute__((ext_vector_type(2)));
typedef float v8f __attribute__((ext_vector_type(8)));

__device__ __forceinline__ v8f wmma_f32_4(v2f a, v2f b, v8f c) {
  // (neg_a, A, neg_b, B, c_mod, C, reuse_a, reuse_b)
  return __builtin_amdgcn_wmma_f32_16x16x4_f32(false, a, false, b, (short)0, c,
                                               false, false);
}

__device__ __forceinline__ float relu_(float x) { return fmaxf(x, 0.0f); }

// ---------------------------------------------------------------------------
// Kernel 1: template path. One workgroup (8 waves) per batch.
//   t_pre[64][196] = w_t[64][256] @ template[b][256][196]   (WMMA)
//   GroupNorm(32 groups) + ReLU in LDS, then 2x2 pool -> t_kernel[7][7]
//   and global mean -> t_global.
// Padding columns (n>=196) of the LDS tile are never read downstream, so the
// B fetch just clamps the column (garbage lands in dead columns).
// ---------------------------------------------------------------------------
__global__ __launch_bounds__(256) void k_template_path(
    const float* __restrict__ tf, const float* __restrict__ w_t,
    const float* __restrict__ gnw, const float* __restrict__ gnb,
    float* __restrict__ t_global, float* __restrict__ t_kernel) {
  const int b = blockIdx.x;
  const int tid = threadIdx.x;
  const int wave = tid >> 5, lane = tid & 31;
  const int hv = lane >> 4, l16 = lane & 15;

  __shared__ float lds_t[64][208];  // 52 KB, padded N
  __shared__ float ga[64], gb[64];

  const float* X = tf + (size_t)b * 256 * 196;

  // --- GEMM: 4 M-tiles x 13 N-tiles = 52 tiles over 8 waves ---
  for (int t = wave; t < 52; t += 8) {
    const int m0 = (t & 3) << 4;
    const int n0 = (t >> 2) << 4;
    const int n = n0 + l16;
    const int nc = (n < 196) ? n : 195;  // clamped, always-valid column
    v8f acc = {};
    for (int k0 = 0; k0 < 256; k0 += 4) {
      const int kk = k0 + (hv << 1);
      v2f a, bb;
      a.x = w_t[(m0 + l16) * 256 + kk];
      a.y = w_t[(m0 + l16) * 256 + kk + 1];
      bb.x = X[kk * 196 + nc];
      bb.y = X[(kk + 1) * 196 + nc];
      acc = wmma_f32_4(a, bb, acc);
    }
#pragma unroll
    for (int r = 0; r < 8; ++r)
      lds_t[m0 + r + (hv << 3)][n0 + l16] = acc[r];
  }
  __syncthreads();

  // --- GroupNorm stats: group g (2 channels x 196 px) by 8 threads ---
  {
    const int g = tid >> 3, l8 = tid & 7;
    float s = 0.f, q = 0.f;
    for (int i = l8; i < 392; i += 8) {
      const float v = lds_t[(g << 1) + (i >= 196)][i % 196];
      s += v;
      q += v * v;
    }
    s += __shfl_xor(s, 1); q += __shfl_xor(q, 1);
    s += __shfl_xor(s, 2); q += __shfl_xor(q, 2);
    s += __shfl_xor(s, 4); q += __shfl_xor(q, 4);
    const float mean = s * (1.0f / 392.0f);
    const float var = q * (1.0f / 392.0f) - mean * mean;
    const float sc = rsqrtf(var + 1e-5f);
    if (l8 < 2) {
      const int c = (g << 1) + l8;
      ga[c] = gnw[c] * sc;
      gb[c] = gnb[c] - mean * sc * gnw[c];
    }
  }
  __syncthreads();

  // --- fused ReLU + 2x2 avg pool -> t_kernel, and global mean ---
  for (int idx = tid; idx < 64 * 49; idx += 256) {
    const int c = idx / 49, kid = idx % 49;
    const int ky = kid / 7, kx = kid % 7;
    const float a = ga[c], bo = gb[c];
    float s = 0.f;
#pragma unroll
    for (int dy = 0; dy < 2; ++dy)
#pragma unroll
      for (int dx = 0; dx < 2; ++dx)
        s += relu_(a * lds_t[c][(2 * ky + dy) * 14 + (2 * kx + dx)] + bo);
    t_kernel[((size_t)b * 64 + c) * 49 + kid] = s * 0.25f;
  }
  if (tid < 64) {
    const int c = tid;
    const float a = ga[c], bo = gb[c];
    float s = 0.f;
    for (int p = 0; p < 196; ++p) s += relu_(a * lds_t[c][p] + bo);
    t_global[(size_t)b * 64 + c] = s * (1.0f / 196.0f);
  }
}

// ---------------------------------------------------------------------------
// Kernel 2: search conv1x1 as WMMA GEMM.
//   s_pre[b][64][4096] = w_s[64][256] @ search[b][256][4096]
// Block: 8 waves; wave owns 1 M-tile x 4 N-tiles (A fragment reused 4x).
// Grid: (4096/128, B).  All loads unconditional.
// ---------------------------------------------------------------------------
__global__ __launch_bounds__(256) void k_search_gemm(
    const float* __restrict__ sf, const float* __restrict__ w_s,
    float* __restrict__ s_pre) {
  const int b = blockIdx.y;
  const int tid = threadIdx.x;
  const int wave = tid >> 5, lane = tid & 31;
  const int hv = lane >> 4, l16 = lane & 15;
  const int m0 = (wave & 3) << 4;
  const int nbase = blockIdx.x * 128 + ((wave >> 2) << 6);

  const float* X = sf + (size_t)b * 256 * 4096;
  v8f acc0 = {}, acc1 = {}, acc2 = {}, acc3 = {};

  for (int k0 = 0; k0 < 256; k0 += 4) {
    const int kk = k0 + (hv << 1);
    v2f a;
    a.x = w_s[(m0 + l16) * 256 + kk];
    a.y = w_s[(m0 + l16) * 256 + kk + 1];
    const float* r0 = X + (size_t)kk * 4096 + nbase + l16;
    const float* r1 = r0 + 4096;
    v2f b0; b0.x = r0[0];  b0.y = r1[0];  acc0 = wmma_f32_4(a, b0, acc0);
    v2f b1; b1.x = r0[16]; b1.y = r1[16]; acc1 = wmma_f32_4(a, b1, acc1);
    v2f b2; b2.x = r0[32]; b2.y = r1[32]; acc2 = wmma_f32_4(a, b2, acc2);
    v2f b3; b3.x = r0[48]; b3.y = r1[48]; acc3 = wmma_f32_4(a, b3, acc3);
  }

  float* O = s_pre + (size_t)b * 64 * 4096;
#pragma unroll
  for (int r = 0; r < 8; ++r) {
    const int row = m0 + r + (hv << 3);
    float* o = O + (size_t)row * 4096 + nbase + l16;
    o[0] = acc0[r];
    o[16] = acc1[r];
    o[32] = acc2[r];
    o[48] = acc3[r];
  }
}

// ---------------------------------------------------------------------------
// Kernel 3: GroupNorm statistics -> per-channel affine (a, b).
// One block per (batch, group): reduces 2 channels x 4096 px.
// ---------------------------------------------------------------------------
__global__ __launch_bounds__(256) void k_gn_stats(
    const float* __restrict__ pre, const float* __restrict__ gnw,
    const float* __restrict__ gnb, float* __restrict__ A,
    float* __restrict__ Bo) {
  const int b = blockIdx.x >> 5, g = blockIdx.x & 31;
  const float* p = pre + ((size_t)b * 64 + g * 2) * 4096;
  const int tid = threadIdx.x;
  float s = 0.f, q = 0.f;
  for (int i = tid; i < 8192; i += 256) {
    const float v = p[i];
    s += v;
    q += v * v;
  }
  __shared__ float ss[256], sq[256];
  ss[tid] = s;
  sq[tid] = q;
  __syncthreads();
  for (int o = 128; o > 0; o >>= 1) {
    if (tid < o) {
      ss[tid] += ss[tid + o];
      sq[tid] += sq[tid + o];
    }
    __syncthreads();
  }
  if (tid < 2) {
    const float mean = ss[0] * (1.f / 8192.f);
    const float var = sq[0] * (1.f / 8192.f) - mean * mean;
    const float sc = rsqrtf(var + 1e-5f);
    const int c = g * 2 + tid;
    A[(size_t)b * 64 + c] = gnw[c] * sc;
    Bo[(size_t)b * 64 + c] = gnb[c] - mean * sc * gnw[c];
  }
}

// ---------------------------------------------------------------------------
// Kernel 4: corr = global_corr + depthwise 7x7.  One block per (b,c).
// 70x70 LDS tile holds relu(GN(s)) with zero halo; 49-tap FMA per pixel.
// Halo fill: clamped-address load + cndmask select (no EXEC branching).
// ---------------------------------------------------------------------------
__global__ __launch_bounds__(256) void k_corr(
    const float* __restrict__ s_pre, const float* __restrict__ sA,
    const float* __restrict__ sB, const float* __restrict__ t_global,
    const float* __restrict__ t_kernel, float* __restrict__ corr) {
  const int bc = blockIdx.x;  // b*64 + c
  const float* src = s_pre + (size_t)bc * 4096;
  const float a = sA[bc], bo = sB[bc], tg = t_global[bc];
  __shared__ float tile[70][70];
  __shared__ float kern[49];
  const int tid = threadIdx.x;
  if (tid < 49) kern[tid] = t_kernel[(size_t)bc * 49 + tid];
  for (int i = tid; i < 70 * 70; i += 256) {
    const int y = i / 70 - 3, x = i % 70 - 3;
    const bool in = ((unsigned)y < 64u) & ((unsigned)x < 64u);
    const float raw = src[in ? (y * 64 + x) : 0];  // unconditional load
    tile[i / 70][i % 70] = in ? relu_(a * raw + bo) : 0.f;
  }
  __syncthreads();
  float* dst = corr + (size_t)bc * 4096;
  for (int p = tid; p < 4096; p += 256) {
    const int y = p >> 6, x = p & 63;
    float acc = tg * tile[y + 3][x + 3];
#pragma unroll
    for (int ky = 0; ky < 7; ++ky)
#pragma unroll
      for (int kx = 0; kx < 7; ++kx)
        acc = fmaf(kern[ky * 7 + kx], tile[y + ky][x + kx], acc);
    dst[p] = acc;
  }
}

// ---------------------------------------------------------------------------
// Kernel 5: 3x3 conv as implicit WMMA GEMM, K = 64*9 = 576.
// A = w_p1 viewed as [64][576] (memory order O,C,3,3 matches k=c*9+dy*3+dx).
// B = im2col of corr gathered from L2 (corr is L2-resident: 32 MB << 192 MB).
// Boundary zeros via clamped-offset load + cndmask (loads unconditional).
// ---------------------------------------------------------------------------
__global__ __launch_bounds__(256) void k_conv3x3_gemm(
    const float* __restrict__ corr, const float* __restrict__ w_p1,
    float* __restrict__ y_pre) {
  const int b = blockIdx.y;
  const int tid = threadIdx.x;
  const int wave = tid >> 5, lane = tid & 31;
  const int hv = lane >> 4, l16 = lane & 15;
  const int m0 = (wave & 3) << 4;
  const int nbase = blockIdx.x * 128 + ((wave >> 2) << 6);

  const float* X = corr + (size_t)b * 64 * 4096;
  v8f acc[4] = {};

  for (int k0 = 0; k0 < 576; k0 += 4) {
    const int kk = k0 + (hv << 1);
    v2f a;
    a.x = w_p1[(m0 + l16) * 576 + kk];
    a.y = w_p1[(m0 + l16) * 576 + kk + 1];

    const int c0 = kk / 9, r0 = kk - c0 * 9;
    const int dy0 = r0 / 3 - 1, dx0 = r0 % 3 - 1;
    const int kk1 = kk + 1;
    const int c1 = kk1 / 9, r1 = kk1 - c1 * 9;
    const int dy1 = r1 / 3 - 1, dx1 = r1 % 3 - 1;
    const float* P0 = X + (size_t)c0 * 4096;
    const float* P1 = X + (size_t)c1 * 4096;

#pragma unroll
    for (int j = 0; j < 4; ++j) {
      const int n = nbase + (j << 4) + l16;
      const int y = n >> 6, x = n & 63;
      const int y0 = y + dy0, x0 = x + dx0;
      const int y1 = y + dy1, x1 = x + dx1;
      const bool in0 = ((unsigned)y0 < 64u) & ((unsigned)x0 < 64u);
      const bool in1 = ((unsigned)y1 < 64u) & ((unsigned)x1 < 64u);
      const float v0 = P0[in0 ? (y0 * 64 + x0) : 0];  // unconditional load
      const float v1 = P1[in1 ? (y1 * 64 + x1) : 0];  // unconditional load
      v2f bb;
      bb.x = in0 ? v0 : 0.f;
      bb.y = in1 ? v1 : 0.f;
      acc[j] = wmma_f32_4(a, bb, acc[j]);
    }
  }

  float* O = y_pre + (size_t)b * 64 * 4096;
#pragma unroll
  for (int r = 0; r < 8; ++r) {
    const int row = m0 + r + (hv << 3);
    float* o = O + (size_t)row * 4096 + nbase + l16;
    o[0] = acc[0][r];
    o[16] = acc[1][r];
    o[32] = acc[2][r];
    o[48] = acc[3][r];
  }
}

// ---------------------------------------------------------------------------
// Kernel 6: fused GN + ReLU + 1x1 (64 -> 1) + bias.
// ---------------------------------------------------------------------------
__global__ __launch_bounds__(256) void k_final(
    const float* __restrict__ y_pre, const float* __restrict__ yA,
    const float* __restrict__ yB, const float* __restrict__ w_p2,
    const float* __restrict__ b_p2, float* __restrict__ out) {
  const int b = blockIdx.x >> 4;
  const int p = ((blockIdx.x & 15) << 8) + threadIdx.x;
  const float* Y = y_pre + (size_t)b * 64 * 4096 + p;
  float acc = b_p2[0];
#pragma unroll 8
  for (int c = 0; c < 64; ++c) {
    const float v = relu_(yA[b * 64 + c] * Y[(size_t)c * 4096] + yB[b * 64 + c]);
    acc = fmaf(w_p2[c], v, acc);
  }
  out[(size_t)b * 4096 + p] = acc;
}

// ---------------------------------------------------------------------------
extern "C" void kernel_launch(void* const* d_in, const int* in_sizes, int n_in,
                              void* d_out, int out_size, void* d_ws,
                              size_t ws_size, hipStream_t stream) {
  (void)in_sizes; (void)n_in; (void)out_size; (void)ws_size;
  const float* tf   = (const float*)d_in[0];   // [32,256,14,14]
  const float* sf   = (const float*)d_in[1];   // [32,256,64,64]
  const float* w_t  = (const float*)d_in[2];   // [64,256]
  const float* gtw  = (const float*)d_in[3];
  const float* gtb  = (const float*)d_in[4];
  const float* w_s  = (const float*)d_in[5];   // [64,256]
  const float* gsw  = (const float*)d_in[6];
  const float* gsb  = (const float*)d_in[7];
  const float* w_p1 = (const float*)d_in[8];   // [64,64,3,3]
  const float* gpw  = (const float*)d_in[9];
  const float* gpb  = (const float*)d_in[10];
  const float* w_p2 = (const float*)d_in[11];  // [1,64]
  const float* b_p2 = (const float*)d_in[12];  // [1]
  float* out = (float*)d_out;                  // [32,1,64,64]

  // workspace layout (floats); bufA/bufB are 32 MB each -> L2 resident
  float* ws = (float*)d_ws;
  float* t_global = ws;                 //  2048
  float* t_kernel = ws + 2048;          // 100352
  float* sA = ws + 102400;              //  2048
  float* sB = ws + 104448;              //  2048
  float* yA = ws + 106496;              //  2048
  float* yB = ws + 108544;              //  2048
  float* bufA = ws + 131072;            // 8388608 (s_pre, later y_pre)
  float* bufB = bufA + 8388608;         // 8388608 (corr)

  k_template_path<<<32, 256, 0, stream>>>(tf, w_t, gtw, gtb, t_global, t_kernel);
  k_search_gemm<<<dim3(32, 32), 256, 0, stream>>>(sf, w_s, bufA);
  k_gn_stats<<<1024, 256, 0, stream>>>(bufA, gsw, gsb, sA, sB);
  k_corr<<<2048, 256, 0, stream>>>(bufA, sA, sB, t_global, t_kernel, bufB);
  k_conv3x3_gemm<<<dim3(32, 32), 256, 0, stream>>>(bufB, w_p1, bufA);
  k_gn_stats<<<1024, 256, 0, stream>>>(bufA, gpw, gpb, yA, yB);
  k_final<<<512, 256, 0, stream>>>(bufA, yA, yB, w_p2, b_p2, out);
}